// EuclideanLoss_25658134626535
// MI455X (gfx1250) — compile-verified
//
#include <hip/hip_runtime.h>

// EuclideanLoss on MI455X (gfx1250, wave32).
// loss = (1/(B*N)) * sum_{b,i} w_i * sqrt( sum_d (x[b,i,d]-y[b,d,i])^2 ),
// w_i = 1.5 for i in {1,2}, else 1.0.
//
// HBM-bound: 128 MiB @ 23.3 TB/s ~= 5.8 us floor. The per-tile sum-of-squares
// is computed as diag(Z*Z^T) with V_WMMA_F32_16X16X4_F32 (fp32 matrix pipe,
// compute is free at this arithmetic intensity). For the symmetric product the
// A-fragment and B-fragment of Z/Z^T are identical, so one v2f fragment feeds
// both WMMA operands.

typedef float v2f __attribute__((ext_vector_type(2)));
typedef float v8f __attribute__((ext_vector_type(8)));

#define B_  32
#define N_  8192
#define D_  64
#define TILES_PER_B   (N_ / 16)                 // 512
#define TILES         (B_ * TILES_PER_B)        // 16384
#define WAVES_PER_BLK 8
#define BLK_THREADS   (WAVES_PER_BLK * 32)      // 256
#define GRID_BLOCKS   (TILES / WAVES_PER_BLK)   // 2048

__global__ __launch_bounds__(BLK_THREADS)
void euclid_tile_wmma(const float* __restrict__ x,   // [B, N, D]
                      const float* __restrict__ y,   // [B, D, N]
                      float* __restrict__ partials)  // [GRID_BLOCKS]
{
    const int lane = threadIdx.x & 31;
    const int wave = threadIdx.x >> 5;
    const int tile = blockIdx.x * WAVES_PER_BLK + wave;   // grid exactly covers TILES

    const int b  = tile >> 9;             // tile / 512
    const int i0 = (tile & 511) << 4;     // 16-row tile start in N
    const int m  = lane & 15;             // row within tile this lane serves
    const int hi = lane >> 4;             // half-wave: K offset 0 or 2

    // Lane L<16 covers row i0+L at d = 4k, 4k+1; lane L+16 covers d = 4k+2, 4k+3.
    const float* xrow = x + (size_t)(b * N_ + i0 + m) * D_;
    const float* ycol = y + (size_t)b * D_ * N_ + (i0 + m);

    v8f acc = {};
    #pragma unroll
    for (int kk = 0; kk < 16; ++kk) {
        const int c0 = 4 * kk + 2 * hi;
        // x: per-lane b64 (row fully consumed across the kk loop -> 1x traffic)
        v2f xv = *(const v2f*)(xrow + c0);
        // y: lanes 0-15 read 64B of row c0, lanes 16-31 of row c0+2 -> coalesced
        float y0 = ycol[(size_t)c0 * N_];
        float y1 = ycol[(size_t)(c0 + 1) * N_];

        v2f a;                     // A (and B) fragment of Z = x_tile - y_tile^T
        a.x = xv.x - y0;
        a.y = xv.y - y1;

        // acc = Z_chunk * Z_chunk^T + acc   (fp32 WMMA, 8 args)
        acc = __builtin_amdgcn_wmma_f32_16x16x4_f32(
                  /*neg_a=*/false, a, /*neg_b=*/false, a,
                  /*c_mod=*/(short)0, acc, /*reuse_a=*/false, /*reuse_b=*/false);
    }

    // diag(Z Z^T): C/D layout -> (M=r, N=lane) lanes 0-15 ; (M=8+r, N=lane-16) lanes 16-31.
    // Diagonal m=0..7 sits at lane m in acc[m]; m=8..15 at lane m+16 in acc[m-8].
    const bool low  = lane < 8;
    const bool high = lane >= 24;
    int r = (low ? lane : (lane - 24)) & 7;
    float val = acc[0];
    if (r == 1) val = acc[1];
    if (r == 2) val = acc[2];
    if (r == 3) val = acc[3];
    if (r == 4) val = acc[4];
    if (r == 5) val = acc[5];
    if (r == 6) val = acc[6];
    if (r == 7) val = acc[7];

    float contrib = 0.0f;
    if (low || high) {
        const int i = i0 + (low ? lane : (lane - 16));
        const float w = (i == 1 || i == 2) ? 1.5f : 1.0f;
        contrib = w * sqrtf(val) * (1.0f / ((float)B_ * (float)N_));
    }

    // wave32 reduction
    #pragma unroll
    for (int off = 16; off > 0; off >>= 1)
        contrib += __shfl_down(contrib, off, 32);

    __shared__ float wsum[WAVES_PER_BLK];
    if (lane == 0) wsum[wave] = contrib;
    __syncthreads();
    if (threadIdx.x == 0) {
        float s = 0.0f;
        #pragma unroll
        for (int j = 0; j < WAVES_PER_BLK; ++j) s += wsum[j];
        partials[blockIdx.x] = s;
    }
}

__global__ __launch_bounds__(256)
void euclid_final_reduce(const float* __restrict__ partials, float* __restrict__ out)
{
    // Deterministic fixed-order reduction of GRID_BLOCKS partials.
    float s = 0.0f;
    for (int j = threadIdx.x; j < GRID_BLOCKS; j += 256) s += partials[j];
    #pragma unroll
    for (int off = 16; off > 0; off >>= 1)
        s += __shfl_down(s, off, 32);

    __shared__ float wsum[8];
    const int lane = threadIdx.x & 31;
    const int wave = threadIdx.x >> 5;
    if (lane == 0) wsum[wave] = s;
    __syncthreads();
    if (threadIdx.x == 0) {
        float t = 0.0f;
        #pragma unroll
        for (int j = 0; j < 8; ++j) t += wsum[j];
        out[0] = t;
    }
}

extern "C" void kernel_launch(void* const* d_in, const int* in_sizes, int n_in,
                              void* d_out, int out_size, void* d_ws, size_t ws_size,
                              hipStream_t stream) {
    const float* x = (const float*)d_in[0];   // [32, 8192, 64] f32
    const float* y = (const float*)d_in[1];   // [32, 64, 8192] f32
    float* out      = (float*)d_out;          // scalar loss
    float* partials = (float*)d_ws;           // GRID_BLOCKS floats (8 KB)

    euclid_tile_wmma<<<GRID_BLOCKS, BLK_THREADS, 0, stream>>>(x, y, partials);
    euclid_final_reduce<<<1, 256, 0, stream>>>(partials, out);
}